// SequenceNLLLoss_87290915324014
// MI455X (gfx1250) — compile-verified
//
#include <hip/hip_runtime.h>

#define EOS_TOKEN 1
#define B_ 8
#define T_ 1024
#define V_ 32000

// One workgroup (1024 threads = 32 wave32 waves) per sequence b.
// Phase 1: first index t with target[b,t]==EOS via per-wave ballot + LDS min.
// Phase 2: masked gather of -logp (non-temporal: zero reuse) + hierarchical sum.
// Writes per-sequence result to ws[b]; a fixed-order finalize kernel sums them
// so the FP result is bit-deterministic across replays.
__global__ __launch_bounds__(1024) void seqnll_per_seq_kernel(
    const float* __restrict__ inp,   // [B, T, V] log-probs
    const int*   __restrict__ tgt,   // [B, T]
    float*       __restrict__ ws)    // [B] per-sequence results
{
    const int b    = blockIdx.x;
    const int t    = threadIdx.x;          // == time index, blockDim.x == T_
    const int wave = t >> 5;
    const int lane = t & 31;

    const int g = tgt[b * T_ + t];

    // ---- Phase 1: first t where target == EOS (reference argmax semantics) ----
    __shared__ int s_min[32];
    __shared__ int s_eos;

    const unsigned long long bal = __ballot(g == EOS_TOKEN);   // wave32: low 32 bits
    int cand = bal ? (wave * 32 + (__ffsll((long long)bal) - 1)) : T_;
    if (lane == 0) s_min[wave] = cand;
    __syncthreads();
    if (wave == 0) {
        int v = s_min[lane];
        #pragma unroll
        for (int off = 16; off > 0; off >>= 1)
            v = min(v, __shfl_down(v, off, 32));
        if (lane == 0) s_eos = v;
    }
    __syncthreads();
    const int eos_idx = s_eos;

    // ---- Phase 2: masked NLL gather + block sum ----
    float nll = 0.0f;
    if (t <= eos_idx) {
        const float* p = inp + ((size_t)b * T_ + (size_t)t) * (size_t)V_ + (size_t)g;
        nll = -__builtin_nontemporal_load(p);   // TH=NT: scattered lines, no reuse
    }
    #pragma unroll
    for (int off = 16; off > 0; off >>= 1)
        nll += __shfl_down(nll, off, 32);

    __shared__ float s_sum[32];
    if (lane == 0) s_sum[wave] = nll;
    __syncthreads();
    if (wave == 0) {
        float v = s_sum[lane];
        #pragma unroll
        for (int off = 16; off > 0; off >>= 1)
            v += __shfl_down(v, off, 32);
        if (lane == 0)
            ws[b] = v / (float)eos_idx;   // reference divides by the index itself
    }
}

// Single-wave fixed-order finalize: deterministic sum of 8 per-seq values.
__global__ __launch_bounds__(32) void seqnll_finalize_kernel(
    const float* __restrict__ ws, float* __restrict__ out)
{
    if (threadIdx.x == 0) {
        float acc = 0.0f;
        #pragma unroll
        for (int i = 0; i < B_; ++i) acc += ws[i];
        out[0] = acc;
    }
}

extern "C" void kernel_launch(void* const* d_in, const int* in_sizes, int n_in,
                              void* d_out, int out_size, void* d_ws, size_t ws_size,
                              hipStream_t stream) {
    const float* inp = (const float*)d_in[0];   // [B, T, V] float32 log-probs
    const int*   tgt = (const int*)d_in[1];     // [B, T] int32
    float*       out = (float*)d_out;           // scalar
    float*       ws  = (float*)d_ws;            // >= B_ floats

    (void)in_sizes; (void)n_in; (void)out_size; (void)ws_size;

    seqnll_per_seq_kernel<<<B_, T_, 0, stream>>>(inp, tgt, ws);
    seqnll_finalize_kernel<<<1, 32, 0, stream>>>(ws, out);
}